// SRN_LSTM_13340168421548
// MI455X (gfx1250) — compile-verified
//
#include <hip/hip_runtime.h>
#include <hip/hip_bf16.h>

// Problem dims (fixed by the reference)
#define B_ 128
#define S_ 512
#define I_ 256
#define H_ 256
#define G_ 1024   // 4*H

typedef float v2f __attribute__((ext_vector_type(2)));
typedef float v8f __attribute__((ext_vector_type(8)));

// Branchless activations: v_exp_f32 + v_rcp_f32, no saveexec divergence.
__device__ __forceinline__ float fast_sigmoid(float v) {
    return __builtin_amdgcn_rcpf(1.0f + __expf(-v));
}
__device__ __forceinline__ float fast_tanh(float v) {
    // tanh(x) = 1 - 2/(exp(2x)+1); saturates correctly to +/-1 via exp->inf/0
    return 1.0f - 2.0f * __builtin_amdgcn_rcpf(1.0f + __expf(2.0f * v));
}

// ---------------------------------------------------------------------------
// Kernel 1: actual_W / actual_U = W + signed-masked noise (elementwise)
// ---------------------------------------------------------------------------
__global__ void srn_prep_weights(
    const float* __restrict__ W,  const float* __restrict__ upW,
    const float* __restrict__ nzW, const float* __restrict__ rdW,
    const float* __restrict__ U,  const float* __restrict__ upU,
    const float* __restrict__ nzU, const float* __restrict__ rdU,
    float* __restrict__ aW, float* __restrict__ aU, int n)
{
    int i = blockIdx.x * blockDim.x + threadIdx.x;
    int stride = gridDim.x * blockDim.x;
    for (; i < n; i += stride) {
        {
            float u = upW[i], nz = nzW[i], r = rdW[i];
            float z = (u / (nz + 1e-8f) + 1.0f) * 0.5f;
            float m = fminf(fmaxf(floorf(z + r), 0.0f), 1.0f) * 2.0f - 1.0f;
            aW[i] = W[i] + nz * m;
        }
        {
            float u = upU[i], nz = nzU[i], r = rdU[i];
            float z = (u / (nz + 1e-8f) + 1.0f) * 0.5f;
            float m = fminf(fmaxf(floorf(z + r), 0.0f), 1.0f) * 2.0f - 1.0f;
            aU[i] = U[i] + nz * m;
        }
    }
}

// ---------------------------------------------------------------------------
// Kernel 2: zero h-ping-pong buffers + grid-barrier counter (contiguous)
// ---------------------------------------------------------------------------
__global__ void srn_zero(float* __restrict__ p, int n)
{
    int i = blockIdx.x * blockDim.x + threadIdx.x;
    int stride = gridDim.x * blockDim.x;
    for (; i < n; i += stride) p[i] = 0.0f;
}

// ---------------------------------------------------------------------------
// Kernel 3: persistent LSTM scan. Grid = 16 WGs (one per 16-wide hidden-col
// block), 256 threads = 8 waves (wave w owns batch rows [16w,16w+16)).
//
// Per step s (single kernel, device-scope grid barrier between steps):
//   gates = x_t @ aW + h_prev @ aU   via V_WMMA_F32_16X16X4_F32
//   i/f/g/o -> c (VGPR-resident all 512 steps), h -> hseq + ping-pong buffer.
// h ping-pong is pure offset arithmetic on one base pointer so the A-loads
// stay in the GLOBAL address space (global_load_b64, not flat_load_b64).
//
// LDS weight layout (per WG, its 64 cols = 4 gate chunks x 16):
//   float2 at index (k/4)*128 + col*2 + (kk&1) holds K-pair (2kk, 2kk+1).
//   Lane (half,l16), tile q reads idx = (k/4)*128 + q*32 + l16*2 + half
//   -> dword bank = l16*4 + half*2 : all 64 banks exactly once, conflict-free
//   ds_load_b64 (compiler pairs them into ds_load_2addr_b64).
//   2 x 64 KB = 128 KB LDS per WG (<= 320 KB WGP LDS).
//
// WMMA f32 fragment layouts per ISA 7.12.2:
//   A 16x4 : lane<16 -> row=lane,{K=k,k+1}; lane>=16 -> row=lane-16,{K=k+2,k+3}
//   B 4x16 : mirrored;  C/D: col = lane&15, row = vgpr + 8*(lane>=16)
// All step-loop branches uniform -> EXEC all-ones at every WMMA.
// ---------------------------------------------------------------------------
__global__ __launch_bounds__(256)
void srn_lstm_scan(const float* __restrict__ x,     // [B,S,I]
                   const float* __restrict__ aW,    // [I,G]
                   const float* __restrict__ aU,    // [H,G]
                   float* __restrict__ hbase,       // [2][B,H] ping-pong
                   float* __restrict__ out,         // hseq | h_T | c_T
                   unsigned* __restrict__ barrier_ctr)
{
    __shared__ v2f wlds[8192];   // 64 KB: actual_W cols for this WG
    __shared__ v2f ulds[8192];   // 64 KB: actual_U cols for this WG

    const int tid  = threadIdx.x;
    const int lane = tid & 31;
    const int wave = tid >> 5;               // 0..7
    const int l16  = lane & 15;
    const int half = lane >> 4;              // 0/1
    const int koff = half * 2;
    const int m0   = wave * 16;              // batch row base (covers all 128)
    const int hc0  = blockIdx.x * 16;        // hidden col base

    // ---- stage this WG's weight columns into LDS (once per launch) ----
    for (int e = tid; e < 8192; e += 256) {
        const int kkpair = e >> 7;           // 0..63
        const int rem    = e & 127;
        const int col    = rem >> 1;         // 0..63
        const int par    = rem & 1;
        const int kk     = kkpair * 2 + par; // K-pair index 0..127
        const int gcol   = (col >> 4) * H_ + hc0 + (col & 15);
        v2f w, u;
        w.x = aW[(size_t)(2 * kk)     * G_ + gcol];
        w.y = aW[(size_t)(2 * kk + 1) * G_ + gcol];
        u.x = aU[(size_t)(2 * kk)     * G_ + gcol];
        u.y = aU[(size_t)(2 * kk + 1) * G_ + gcol];
        wlds[e] = w;
        ulds[e] = u;
    }
    __syncthreads();

    float cc[8] = {0.f, 0.f, 0.f, 0.f, 0.f, 0.f, 0.f, 0.f};  // cell state (VGPRs)
    const float*  xbase   = x + (size_t)(m0 + l16) * S_ * I_;
    const size_t  hstride = (size_t)B_ * H_;
    const int     col     = hc0 + l16;
    unsigned      target  = 0;

    for (int s = 0; s < S_; ++s) {
        const float* hp = hbase + (size_t)(s & 1) * hstride;       // read
        float*       hn = hbase + (size_t)((s + 1) & 1) * hstride; // write

        v8f acc[4] = {v8f{}, v8f{}, v8f{}, v8f{}};

        // ---- x_t @ aW : K = 0..255 ----
        const float* xrow = xbase + (size_t)s * I_;
        #pragma unroll 2
        for (int k = 0; k < I_; k += 4) {
            v2f a = *(const v2f*)(xrow + k + koff);        // global_load_b64
            const int bidx = (k >> 2) * 128 + l16 * 2 + half;
            #pragma unroll
            for (int q = 0; q < 4; ++q) {
                v2f b = wlds[bidx + q * 32];               // ds_load_b64
                acc[q] = __builtin_amdgcn_wmma_f32_16x16x4_f32(
                    false, a, false, b, (short)0, acc[q], false, false);
            }
        }
        // ---- h_prev @ aU : K = 0..255 ----
        const float* hrow = hp + (size_t)(m0 + l16) * H_;
        #pragma unroll 2
        for (int k = 0; k < H_; k += 4) {
            v2f a = *(const v2f*)(hrow + k + koff);        // global_load_b64
            const int bidx = (k >> 2) * 128 + l16 * 2 + half;
            #pragma unroll
            for (int q = 0; q < 4; ++q) {
                v2f b = ulds[bidx + q * 32];
                acc[q] = __builtin_amdgcn_wmma_f32_16x16x4_f32(
                    false, a, false, b, (short)0, acc[q], false, false);
            }
        }

        // ---- fused, branchless LSTM cell update; c stays in registers ----
        #pragma unroll
        for (int i = 0; i < 8; ++i) {
            const int row = m0 + i + half * 8;   // batch index of this acc row
            const float gi = fast_sigmoid(acc[0][i]);
            const float gf = fast_sigmoid(acc[1][i]);
            const float gg = fast_tanh(acc[2][i]);
            const float go = fast_sigmoid(acc[3][i]);
            const float cn = gf * cc[i] + gi * gg;
            cc[i] = cn;
            const float hv = go * fast_tanh(cn);
            hn[(size_t)row * H_ + col] = hv;
            out[((size_t)row * S_ + s) * H_ + col] = hv;
            if (s == S_ - 1) {                   // uniform branch
                out[(size_t)B_ * S_ * H_ + (size_t)row * H_ + col] = hv;                   // h_T
                out[(size_t)B_ * S_ * H_ + (size_t)B_ * H_ + (size_t)row * H_ + col] = cn; // c_T
            }
        }

        // ---- device-scope grid barrier between timesteps ----
        if (s + 1 < S_) {                        // uniform branch
            __threadfence();                     // release own h stores
            __syncthreads();                     // whole WG done + fenced
            target += gridDim.x;
            if (tid == 0) {
                __hip_atomic_fetch_add(barrier_ctr, 1u, __ATOMIC_RELEASE,
                                       __HIP_MEMORY_SCOPE_AGENT);
                while (__hip_atomic_load(barrier_ctr, __ATOMIC_ACQUIRE,
                                         __HIP_MEMORY_SCOPE_AGENT) < target)
                    __builtin_amdgcn_s_sleep(2);
            }
            __syncthreads();                     // all threads see barrier done
            __threadfence();                     // acquire: invalidate stale h
        }
    }
}

// ---------------------------------------------------------------------------
extern "C" void kernel_launch(void* const* d_in, const int* in_sizes, int n_in,
                              void* d_out, int out_size, void* d_ws, size_t ws_size,
                              hipStream_t stream)
{
    (void)in_sizes; (void)n_in; (void)out_size; (void)ws_size;

    const float* x   = (const float*)d_in[0];
    const float* W   = (const float*)d_in[1];
    const float* U   = (const float*)d_in[2];
    const float* upW = (const float*)d_in[3];
    const float* nzW = (const float*)d_in[4];
    const float* rdW = (const float*)d_in[5];
    const float* upU = (const float*)d_in[6];
    const float* nzU = (const float*)d_in[7];
    const float* rdU = (const float*)d_in[8];
    float* out = (float*)d_out;

    // Workspace (floats): aW | aU | h0 | h1 | ctr   (~2.4 MB)
    float* aW    = (float*)d_ws;
    float* aU    = aW + (size_t)I_ * G_;
    float* hbase = aU + (size_t)H_ * G_;          // h0 then h1, contiguous
    float* ctrf  = hbase + 2 * (size_t)B_ * H_;   // barrier counter
    unsigned* ctr = (unsigned*)ctrf;

    srn_prep_weights<<<256, 256, 0, stream>>>(W, upW, nzW, rdW,
                                              U, upU, nzU, rdU,
                                              aW, aU, I_ * G_);
    // zero h0 (initial state), h1 (harmless), and the barrier counter
    srn_zero<<<64, 256, 0, stream>>>(hbase, 2 * B_ * H_ + 16);

    srn_lstm_scan<<<16, 256, 0, stream>>>(x, aW, aU, hbase, out, ctr);
    // S_ even: h_T/c_T were written from registers at step S_-1 inside the scan.
}